// EmbeddingMul2_16183436772039
// MI455X (gfx1250) — compile-verified
//
#include <hip/hip_runtime.h>
#include <stdint.h>

typedef uint32_t u32;
typedef __attribute__((ext_vector_type(4))) u32 u32x4;
typedef __attribute__((ext_vector_type(8))) u32 u32x8;

#define ROWS_PER_BLOCK 16   // 16-bit gather-index mode: 16 rows per descriptor
#define DIM 256             // floats per embedding row
#define NUM_ENTS 50000

// Launch config guarantees nrows % ROWS_PER_BLOCK == 0 (2048 / 16 = 128 full
// tiles), so every block is a full tile: no tail guards, branch-free prologue.
__global__ __launch_bounds__(32) void emb_gather_tdm_kernel(
    const int* __restrict__ idx, const float* __restrict__ emb,
    float* __restrict__ out)
{
    __shared__ float tile[ROWS_PER_BLOCK * DIM];  // 16 KB staging tile in LDS

    const u32 base = (u32)blockIdx.x * ROWS_PER_BLOCK;

    // ---- 16 row indices: one s_load_b512; pack pairs into 16-bit halves.
    // Indices are < 50000 < 2^16, so upper halves are already zero: no masks.
    const int* ip = idx + base;
    u32 pk[8];
#pragma unroll
    for (int i = 0; i < 8; ++i) {
        u32 lo = (u32)ip[2 * i];
        u32 hi = (u32)ip[2 * i + 1];
        pk[i] = lo | (hi << 16);
    }

    const u32 lds_addr = (u32)(uintptr_t)&tile[0];
    const uint64_t ga  = (uint64_t)(uintptr_t)emb;
    const uint64_t oa  = (uint64_t)(uintptr_t)(out + (size_t)base * DIM);

    // ================= gather-mode load descriptor =================
    u32x4 g0;
    g0.x = 0x80000001u;                       // count=1 | gather_mode=1 | 16-bit idx
    g0.y = lds_addr;                          // lds_addr (bytes)
    g0.z = (u32)ga;                           // global_addr[31:0]
    g0.w = ((u32)(ga >> 32) & 0x01FFFFFFu)    // global_addr[56:32]
         | (2u << 30);                        // type = 2 ("image")

    u32x8 g1;
    g1.s0 = 0x00020000u;                      // wg_mask=0 | data_size=2 (4 bytes)
    g1.s1 = ((u32)DIM & 0xFFFFu) << 16;       // tensor_dim0[15:0]  @ bits 63:48
    g1.s2 = ((u32)DIM >> 16)                  // tensor_dim0[31:16] @ bits 79:64
          | (((u32)NUM_ENTS & 0xFFFFu) << 16);// tensor_dim1[15:0]  @ bits 95:80
    g1.s3 = ((u32)NUM_ENTS >> 16)             // tensor_dim1[31:16] @ bits 111:96
          | (((u32)DIM & 0xFFFFu) << 16);     // tile_dim0 = 256    @ bits 127:112
    g1.s4 = (u32)ROWS_PER_BLOCK;              // tile_dim1 = #valid indices = 16
    g1.s5 = (u32)DIM;                         // tensor_dim0_stride[31:0]
    g1.s6 = 0u;                               // stride hi / dim1_stride (ignored)
    g1.s7 = 0u;

    u32x4 g2; g2.x = pk[0]; g2.y = pk[1]; g2.z = pk[2]; g2.w = pk[3];
    u32x4 g3; g3.x = pk[4]; g3.y = pk[5]; g3.z = pk[6]; g3.w = pk[7];

    asm volatile("tensor_load_to_lds %0, %1, %2, %3"
                 :: "s"(g0), "s"(g1), "s"(g2), "s"(g3) : "memory");
    __builtin_amdgcn_s_wait_tensorcnt(0);     // LDS tile fully written

    // ========== linear store descriptor: 16 contiguous output rows ==========
    enum { TELEMS = ROWS_PER_BLOCK * DIM };   // 4096 elements, 1-D tile

    u32x4 s0;
    s0.x = 0x00000001u;                       // count=1, normal mode
    s0.y = lds_addr;
    s0.z = (u32)oa;
    s0.w = ((u32)(oa >> 32) & 0x01FFFFFFu) | (2u << 30);

    u32x8 s1;
    s1.s0 = 0x00020000u;                      // data_size = 4 bytes
    s1.s1 = ((u32)TELEMS & 0xFFFFu) << 16;    // tensor_dim0[15:0]
    s1.s2 = ((u32)TELEMS >> 16)               // tensor_dim0[31:16]
          | (1u << 16);                       // tensor_dim1 = 1
    s1.s3 = ((u32)TELEMS & 0xFFFFu) << 16;    // tile_dim0 = 4096 (1-D tile)
    s1.s4 = 0u;                               // tile_dim1 = 0 (unused)
    s1.s5 = (u32)TELEMS;                      // tensor_dim0_stride
    s1.s6 = 0u;
    s1.s7 = 0u;

    asm volatile("tensor_store_from_lds %0, %1"
                 :: "s"(s0), "s"(s1) : "memory");
    __builtin_amdgcn_s_wait_tensorcnt(0);
}

extern "C" void kernel_launch(void* const* d_in, const int* in_sizes, int n_in,
                              void* d_out, int out_size, void* d_ws, size_t ws_size,
                              hipStream_t stream) {
    (void)n_in; (void)out_size; (void)d_ws; (void)ws_size;
    const int*   idx = (const int*)d_in[0];      // input_   : 8*256 int32
    const float* emb = (const float*)d_in[1];    // embedding: 50000*256 f32
    float*       out = (float*)d_out;            // out      : 2048*256 f32

    int nrows  = in_sizes[0];                    // 2048 (multiple of 16)
    int blocks = nrows / ROWS_PER_BLOCK;         // 128 full tiles
    emb_gather_tdm_kernel<<<blocks, 32, 0, stream>>>(idx, emb, out);
}